// ResidualConvBlock_63170378990249
// MI455X (gfx1250) — compile-verified
//
#include <hip/hip_runtime.h>

// ---------------- problem constants (fixed by the reference) ----------------
#define N_PTS   65536
#define KNB     32          // neighbors per point
#define P_KP    15          // kernel points
#define CIN     128
#define CM      32          // bottleneck channels
#define SIGMA_F 2.0f
#define EPS_F   1e-5f
#define SHADOW_F 1e6f

typedef float v2f __attribute__((ext_vector_type(2)));
typedef float v8f __attribute__((ext_vector_type(8)));

// exact-fp32 matrix op: D(16x16) = A(16x4) * B(4x16) + C
#define WMMA_F32(A, B, C) \
    __builtin_amdgcn_wmma_f32_16x16x4_f32(false, (A), false, (B), (short)0, (C), false, false)

__device__ __forceinline__ float leaky(float x) { return x >= 0.0f ? x : 0.1f * x; }

// --------------------------------------------------------------------------
// CDNA5 async copy: per-lane 128B global -> LDS, tracked by ASYNCcnt.
// INST_OFFSET is added to BOTH the global and the LDS address (ISA 08 §4.4),
// so the LDS destination row must be contiguous 128B (stride CM = 32 floats).
// --------------------------------------------------------------------------
__device__ __forceinline__ void async_gather_row128(unsigned lds_byte,
                                                    unsigned long long gaddr) {
    asm volatile(
        "global_load_async_to_lds_b128 %0, %1, off\n\t"
        "global_load_async_to_lds_b128 %0, %1, off offset:16\n\t"
        "global_load_async_to_lds_b128 %0, %1, off offset:32\n\t"
        "global_load_async_to_lds_b128 %0, %1, off offset:48\n\t"
        "global_load_async_to_lds_b128 %0, %1, off offset:64\n\t"
        "global_load_async_to_lds_b128 %0, %1, off offset:80\n\t"
        "global_load_async_to_lds_b128 %0, %1, off offset:96\n\t"
        "global_load_async_to_lds_b128 %0, %1, off offset:112"
        :: "v"(lds_byte), "v"(gaddr) : "memory");
}

// ============================================================================
// Kernel 1: x = leaky(LN(s_feats @ w1 + b1)) -> x_pad; also per-row sums rsum
// one wave32 block per 16-point tile; GEMM 16x128x32 via f32 WMMA
// ============================================================================
__global__ void __launch_bounds__(32) k_unary1(
    const float* __restrict__ sf, const float* __restrict__ w1,
    const float* __restrict__ b1, const float* __restrict__ g1,
    const float* __restrict__ bb1, float* __restrict__ xpad,
    float* __restrict__ rsum)
{
    const int L     = threadIdx.x;        // 0..31
    const int pt0   = blockIdx.x * 16;
    const int mrow  = L & 15;             // WMMA M / N lane index
    const int khalf = (L >> 4) * 2;       // K offset of this lane half
    const int rbase = (L >> 4) * 8;       // C/D row base

    __shared__ float xs[16 * 33];         // 16x32 tile, padded stride 33

    if (blockIdx.x == 0) {                // shadow row (index N): zero feats+sum
        xpad[(size_t)N_PTS * CM + L] = 0.0f;
        if (L == 0) rsum[N_PTS] = 0.0f;
    }

    v8f acc0 = {}; v8f acc1 = {};
    for (int k0 = 0; k0 < CIN; k0 += 4) {
        const int k = k0 + khalf;
        v2f a;
        a.x = sf[(size_t)(pt0 + mrow) * CIN + k + 0];
        a.y = sf[(size_t)(pt0 + mrow) * CIN + k + 1];
        v2f b0, b1v;
        b0.x  = w1[(k + 0) * CM + mrow];
        b0.y  = w1[(k + 1) * CM + mrow];
        b1v.x = w1[(k + 0) * CM + 16 + mrow];
        b1v.y = w1[(k + 1) * CM + 16 + mrow];
        acc0 = WMMA_F32(a, b0,  acc0);
        acc1 = WMMA_F32(a, b1v, acc1);
    }
#pragma unroll
    for (int r = 0; r < 8; ++r) {
        const int row = rbase + r;
        xs[row * 33 + mrow]      = acc0[r] + b1[mrow];
        xs[row * 33 + 16 + mrow] = acc1[r] + b1[16 + mrow];
    }
    __syncthreads();
    if (L < 16) {                          // lane L owns point row L
        float s = 0.0f;
        for (int c = 0; c < CM; ++c) s += xs[L * 33 + c];
        const float m = s * (1.0f / CM);
        float v = 0.0f;
        for (int c = 0; c < CM; ++c) { float d = xs[L * 33 + c] - m; v += d * d; }
        const float r = rsqrtf(v * (1.0f / CM) + EPS_F);
        float srow = 0.0f;
        for (int c = 0; c < CM; ++c) {
            float val = (xs[L * 33 + c] - m) * r * g1[c] + bb1[c];
            val = leaky(val);
            xpad[(size_t)(pt0 + L) * CM + c] = val;
            srow += val;
        }
        rsum[pt0 + L] = srow;              // for KPConv valid-neighbor count
    }
}

// ============================================================================
// Kernel 2: KPConv. 64-thread block (2 waves), 16-point tile.
//  Depth-2 software pipeline: async gather (ASYNCcnt) of point m+1 overlaps
//  the einsum1 WMMA of point m. einsum2 batched over the 16-point tile.
// ============================================================================
__global__ void __launch_bounds__(64) k_kpconv(
    const float* __restrict__ xpad, const float* __restrict__ rsum,
    const float* __restrict__ qp,   const float* __restrict__ sp,
    const int*   __restrict__ nbr,  const float* __restrict__ kp,
    const float* __restrict__ kpw,  float* __restrict__ y)
{
    const int tid   = threadIdx.x;      // 0..63
    const int L     = tid & 31;
    const int wv    = tid >> 5;         // wave id 0/1
    const int pt0   = blockIdx.x * 16;
    const int mrow  = L & 15;
    const int khalf = (L >> 4) * 2;
    const int rbase = (L >> 4) * 8;

    __shared__ float kpS[P_KP * 3];
    __shared__ float allwT[2][16 * 33];      // per-wave A = all_w^T (p x k), pad 33
    __shared__ float nbBuf[2][2][KNB * CM];  // per-wave double-buffered 32x32 tile,
                                             // stride 32 (async dest must match global)
    __shared__ float wfAll[16 * 513];        // 16 pts x (15p x 32c); stride 513
    __shared__ float nnInv[16];

    if (tid < P_KP * 3) kpS[tid] = kp[tid];
    __syncthreads();

    float* aw = allwT[wv];
    float wreg[P_KP];
    float rs_cur = 0.0f, rs_next = 0.0f;

    // ---- stage: correlation weights -> wreg, async gather -> nbBuf[wv][buf]
    auto stage_point = [&](int pt, int buf) -> float {
        const int idx = nbr[pt * KNB + L];      // lane L = neighbor k
        float px, py, pz;
        if (idx == N_PTS) { px = py = pz = SHADOW_F; }
        else { px = sp[idx * 3 + 0]; py = sp[idx * 3 + 1]; pz = sp[idx * 3 + 2]; }
        const float qx = qp[pt * 3 + 0], qy = qp[pt * 3 + 1], qz = qp[pt * 3 + 2];
        const float rx = px - qx, ry = py - qy, rz = pz - qz;
#pragma unroll
        for (int p = 0; p < P_KP; ++p) {
            const float dx = rx - kpS[p * 3 + 0];
            const float dy = ry - kpS[p * 3 + 1];
            const float dz = rz - kpS[p * 3 + 2];
            const float d  = sqrtf(dx * dx + dy * dy + dz * dz);
            const float wg = 1.0f - d * (1.0f / SIGMA_F);
            wreg[p] = wg > 0.0f ? wg : 0.0f;
        }
        async_gather_row128((unsigned)(uintptr_t)(&nbBuf[wv][buf][L * CM]),
                            (unsigned long long)(uintptr_t)(xpad + (size_t)idx * CM));
        return rsum[idx];
    };

    rs_cur = stage_point(pt0 + wv * 8, 0);       // prologue: stage point 0

#pragma unroll
    for (int m = 0; m < 8; ++m) {
        const int mt = wv * 8 + m;               // point index within tile
        // 1) commit staged weights for point m into aw
#pragma unroll
        for (int p = 0; p < P_KP; ++p) aw[p * 33 + L] = wreg[p];
        aw[15 * 33 + L] = 0.0f;                  // zero pad row p=15
        // 2) valid-neighbor count for point m
        {
            const unsigned long long ball = __ballot(rs_cur > 0.0f);
            int cnt = __popcll(ball);
            if (cnt < 1) cnt = 1;
            if (L == 0) nnInv[mt] = 1.0f / (float)cnt;
        }
        // 3) stage point m+1 (async gather in flight during m's WMMA)
        if (m < 7) rs_next = stage_point(pt0 + mt + 1, (m + 1) & 1);
        // 4) retire point m's async batch (in-order completion); m+1 stays out
        if (m < 7) asm volatile("s_wait_asynccnt 0x8" ::: "memory");
        else       asm volatile("s_wait_asynccnt 0x0" ::: "memory");
        asm volatile("s_wait_dscnt 0x0" ::: "memory");   // aw stores visible
        // 5) einsum1: wf(16x32) = allwT(16x32) x nbx(32x32), f32 WMMA
        const float* nb = nbBuf[wv][m & 1];
        v8f a0 = {}; v8f a1 = {};
        for (int k0 = 0; k0 < KNB; k0 += 4) {
            const int k = k0 + khalf;
            v2f A;  A.x  = aw[mrow * 33 + k + 0];        A.y  = aw[mrow * 33 + k + 1];
            v2f B0; B0.x = nb[(k + 0) * CM + mrow];      B0.y = nb[(k + 1) * CM + mrow];
            v2f B1; B1.x = nb[(k + 0) * CM + 16 + mrow]; B1.y = nb[(k + 1) * CM + 16 + mrow];
            a0 = WMMA_F32(A, B0, a0);
            a1 = WMMA_F32(A, B1, a1);
        }
#pragma unroll
        for (int r = 0; r < 8; ++r) {
            const int row = rbase + r;               // p index
            wfAll[mt * 513 + row * 32 + mrow]      = a0[r];
            wfAll[mt * 513 + row * 32 + 16 + mrow] = a1[r];
        }
        rs_cur = rs_next;
        asm volatile("" ::: "memory");               // keep pipeline stage order
    }
    __syncthreads();

    // ---- einsum2: out(16x32) = wfAll(16x480) x kpw(480x32); wave wv -> n-tile wv
    v8f acc = {};
    for (int k0 = 0; k0 < P_KP * CM; k0 += 4) {
        const int k = k0 + khalf;                    // k = p*32 + c'
        v2f A; A.x = wfAll[mrow * 513 + k + 0];
               A.y = wfAll[mrow * 513 + k + 1];
        v2f B; B.x = kpw[(k + 0) * CM + wv * 16 + mrow];
               B.y = kpw[(k + 1) * CM + wv * 16 + mrow];
        acc = WMMA_F32(A, B, acc);
    }
#pragma unroll
    for (int r = 0; r < 8; ++r) {
        const int row = rbase + r;                   // point within tile
        y[(size_t)(pt0 + row) * CM + wv * 16 + mrow] = acc[r] * nnInv[row];
    }
}

// ============================================================================
// Kernel 3: x2 = leaky(LN(y)); z = LN(x2 @ w2 + b2); out = leaky(z + s_feats)
// one wave32 block per 16-point tile; GEMM 16x32x128 via f32 WMMA
// ============================================================================
__global__ void __launch_bounds__(32) k_unary2(
    const float* __restrict__ yb, const float* __restrict__ gn,
    const float* __restrict__ bn, const float* __restrict__ w2,
    const float* __restrict__ b2, const float* __restrict__ g2,
    const float* __restrict__ bb2, const float* __restrict__ sf,
    float* __restrict__ outp)
{
    const int L     = threadIdx.x;
    const int pt0   = blockIdx.x * 16;
    const int mrow  = L & 15;
    const int khalf = (L >> 4) * 2;
    const int rbase = (L >> 4) * 8;

    __shared__ float x2[16 * 33];
    __shared__ float z[16 * 129];
    __shared__ float mu[16], rs[16];

    if (L < 16) {                          // leaky(LN(y_row)) over 32 channels
        const float* yr = yb + (size_t)(pt0 + L) * CM;
        float s = 0.0f;
        for (int c = 0; c < CM; ++c) s += yr[c];
        const float m = s * (1.0f / CM);
        float v = 0.0f;
        for (int c = 0; c < CM; ++c) { float d = yr[c] - m; v += d * d; }
        const float r = rsqrtf(v * (1.0f / CM) + EPS_F);
        for (int c = 0; c < CM; ++c) {
            float val = (yr[c] - m) * r * gn[c] + bn[c];
            x2[L * 33 + c] = leaky(val);
        }
    }
    __syncthreads();

    v8f acc[8] = {};
    for (int k0 = 0; k0 < CM; k0 += 4) {
        const int k = k0 + khalf;
        v2f A; A.x = x2[mrow * 33 + k + 0]; A.y = x2[mrow * 33 + k + 1];
#pragma unroll
        for (int nt = 0; nt < 8; ++nt) {
            v2f B; B.x = w2[(k + 0) * CIN + nt * 16 + mrow];
                   B.y = w2[(k + 1) * CIN + nt * 16 + mrow];
            acc[nt] = WMMA_F32(A, B, acc[nt]);
        }
    }
#pragma unroll
    for (int nt = 0; nt < 8; ++nt)
#pragma unroll
        for (int r = 0; r < 8; ++r)
            z[(rbase + r) * 129 + nt * 16 + mrow] = acc[nt][r] + b2[nt * 16 + mrow];
    __syncthreads();

    if (L < 16) {                          // LN stats over 128
        float s = 0.0f;
        for (int c = 0; c < CIN; ++c) s += z[L * 129 + c];
        const float m = s * (1.0f / CIN);
        float v = 0.0f;
        for (int c = 0; c < CIN; ++c) { float d = z[L * 129 + c] - m; v += d * d; }
        mu[L] = m;
        rs[L] = rsqrtf(v * (1.0f / CIN) + EPS_F);
    }
    __syncthreads();
#pragma unroll 4
    for (int i = 0; i < (16 * CIN) / 32; ++i) {
        const int flat = i * 32 + L;
        const int row = flat >> 7, c = flat & 127;
        float val = (z[row * 129 + c] - mu[row]) * rs[row] * g2[c] + bb2[c]
                  + sf[(size_t)(pt0 + row) * CIN + c];
        outp[(size_t)(pt0 + row) * CIN + c] = leaky(val);
    }
}

// ============================================================================
extern "C" void kernel_launch(void* const* d_in, const int* in_sizes, int n_in,
                              void* d_out, int out_size, void* d_ws, size_t ws_size,
                              hipStream_t stream)
{
    const float* s_feats  = (const float*)d_in[0];
    const float* q_points = (const float*)d_in[1];
    const float* s_points = (const float*)d_in[2];
    const int*   nbr      = (const int*)  d_in[3];
    const float* w1       = (const float*)d_in[4];
    const float* b1       = (const float*)d_in[5];
    const float* g1       = (const float*)d_in[6];
    const float* bb1      = (const float*)d_in[7];
    const float* kpw      = (const float*)d_in[8];
    const float* kp       = (const float*)d_in[9];
    const float* gn       = (const float*)d_in[10];
    const float* bn       = (const float*)d_in[11];
    const float* w2       = (const float*)d_in[12];
    const float* b2       = (const float*)d_in[13];
    const float* g2       = (const float*)d_in[14];
    const float* bb2      = (const float*)d_in[15];
    float* outp = (float*)d_out;

    // workspace: x_pad[(N+1) x 32] | y[N x 32] | rsum[N+1]
    float* xpad = (float*)d_ws;
    float* y    = xpad + (size_t)(N_PTS + 1) * CM;
    float* rsum = y + (size_t)N_PTS * CM;

    const int tiles = N_PTS / 16;   // 4096
    k_unary1<<<tiles, 32, 0, stream>>>(s_feats, w1, b1, g1, bb1, xpad, rsum);
    k_kpconv<<<tiles, 64, 0, stream>>>(xpad, rsum, q_points, s_points, nbr, kp, kpw, y);
    k_unary2<<<tiles, 32, 0, stream>>>(y, gn, bn, w2, b2, g2, bb2, s_feats, outp);
}